// F2FBlock_18090402251519
// MI455X (gfx1250) — compile-verified
//
#include <hip/hip_runtime.h>
#include <math.h>

// ---------------------------------------------------------------------------
// GraphSAGE block on MI455X (gfx1250, wave32).
// GEMMs use native fp32 WMMA: V_WMMA_F32_16X16X4_F32.
//   A 16x4  : 2 f32/lane  (lane 0-15: K=0,1 ; lane 16-31: K=2,3), M = lane&15
//   B 4x16  : 2 f32/lane  (mirror of A),                          N = lane&15
//   C/D 16x16: 8 f32/lane (VGPR i: M=i + 8*(lane>>4), N = lane&15)
// One wave computes a 16x128 output tile: 8 accumulators, 32 K-steps.
// Weights are staged once per workgroup into LDS (K-pair float2 layout,
// row length 144 pairs = 288 dwords == 32 mod 64 banks, so the two lane
// halves read disjoint bank sets -> conflict-free ds_load_b64).
// ---------------------------------------------------------------------------

typedef __attribute__((ext_vector_type(2))) float v2f;
typedef __attribute__((ext_vector_type(8))) float v8f;

#define D      128
#define KSTEPS (D / 4)    // 32 WMMA K-steps
#define KPAIRS (D / 2)    // 64 K-pairs staged in LDS
#define NTILES (D / 16)   // 8 column tiles of 16
#define LDSPAD 144        // pair-row length: 288 dwords -> +32 banks per row
#define WAVES  8
#define TPB    (WAVES * 32)

__device__ __forceinline__ float gelu_exact(float v) {
    return 0.5f * v * (1.0f + erff(v * 0.70710678118654752440f));
}

// ---------------------------------------------------------------------------
// zero fill
// ---------------------------------------------------------------------------
__global__ void k_zero(float* __restrict__ p, long n) {
    long i      = (long)blockIdx.x * blockDim.x + threadIdx.x;
    long stride = (long)gridDim.x * blockDim.x;
    for (; i < n; i += stride) p[i] = 0.0f;
}

// ---------------------------------------------------------------------------
// out[r][c] = act( A1[r][:]·W1[c][:] + bias[c] (+ A2[r][:]·W2[c][:]) )
// W row-major [D_out][D_in] (PyTorch layout), i.e. out = A @ W^T + b.
// DUAL is a compile-time switch: no runtime branches in the hot loop.
// ---------------------------------------------------------------------------
template <bool DUAL>
__global__ void __launch_bounds__(TPB)
k_gemm(const float* __restrict__ A1, const float* __restrict__ W1,
       const float* __restrict__ bias,
       const float* __restrict__ A2, const float* __restrict__ W2,
       float* __restrict__ out, int rows, int do_gelu) {
    // LDS-staged weights: lb[kp][col] = {W[col][2kp], W[col][2kp+1]}
    __shared__ v2f lb1[KPAIRS][LDSPAD];
    __shared__ v2f lb2[DUAL ? KPAIRS : 1][LDSPAD];

    const int wave = threadIdx.x >> 5;
    const int lane = threadIdx.x & 31;
    const int half = lane >> 4;   // 0: K={0,1}, 1: K={2,3} within each step
    const int l15  = lane & 15;   // M (for A) / N (for B)

    // Cooperative, coalesced staging: 64 consecutive threads sweep one
    // 512B weight row; a wave's LDS writes (stride 288 dwords) hit only 4 banks.
    for (int idx = threadIdx.x; idx < KPAIRS * D; idx += TPB) {
        const int col = idx >> 6;   // 0..127
        const int kp  = idx & 63;   // 0..63
        lb1[kp][col] = *(const v2f*)(W1 + (col << 7) + (kp << 1));
        if (DUAL) lb2[kp][col] = *(const v2f*)(W2 + (col << 7) + (kp << 1));
    }
    __syncthreads();

    const int tile = blockIdx.x * WAVES + wave;
    const int row0 = tile * 16;
    if (row0 + 16 > rows) return;   // after the barrier: workgroup-safe

    v8f acc[NTILES];
#pragma unroll
    for (int n = 0; n < NTILES; ++n) acc[n] = (v8f){0, 0, 0, 0, 0, 0, 0, 0};

    const float* arow1 = A1 + (long)(row0 + l15) * D;
    const float* arow2 = DUAL ? (A2 + (long)(row0 + l15) * D) : A1;

#pragma unroll 2
    for (int t = 0; t < KSTEPS; ++t) {
        const int k  = t * 4 + half * 2;   // A element offset
        const int kp = t * 2 + half;       // LDS K-pair row
        v2f a1 = *(const v2f*)(arow1 + k);
        v2f a2;
        if (DUAL) a2 = *(const v2f*)(arow2 + k);
#pragma unroll
        for (int n = 0; n < NTILES; ++n) {
            const int col = n * 16 + l15;
            v2f b1 = lb1[kp][col];
            acc[n] = __builtin_amdgcn_wmma_f32_16x16x4_f32(
                false, a1, false, b1, (short)0, acc[n], false, false);
            if (DUAL) {
                v2f b2 = lb2[kp][col];
                acc[n] = __builtin_amdgcn_wmma_f32_16x16x4_f32(
                    false, a2, false, b2, (short)0, acc[n], false, false);
            }
        }
    }

#pragma unroll
    for (int n = 0; n < NTILES; ++n) {
        const int col = n * 16 + l15;
        const float bb = bias[col];
#pragma unroll
        for (int i = 0; i < 8; ++i) {
            const int m = row0 + i + half * 8;
            float v = acc[n][i] + bb;
            if (do_gelu) v = gelu_exact(v);
            out[(long)m * D + col] = v;
        }
    }
}

// ---------------------------------------------------------------------------
// Edge scatter: agg[dst] += h[src]; deg[dst] += 1 (optional).
// One wave per edge: 512B coalesced gather (float4/lane), fp32 atomic scatter.
// ---------------------------------------------------------------------------
__global__ void __launch_bounds__(TPB)
k_scatter(const float* __restrict__ h, const int* __restrict__ src,
          const int* __restrict__ dst, float* __restrict__ agg,
          float* __restrict__ deg, int nedges, int do_deg) {
    const int tid  = blockIdx.x * blockDim.x + threadIdx.x;
    const int e    = tid >> 5;
    const int lane = tid & 31;
    if (e >= nedges) return;
    const int s = src[e];
    const int d = dst[e];
    const float4 v = ((const float4*)(h + (long)s * D))[lane];
    float* ap = agg + (long)d * D + lane * 4;
    atomicAdd(ap + 0, v.x);
    atomicAdd(ap + 1, v.y);
    atomicAdd(ap + 2, v.z);
    atomicAdd(ap + 3, v.w);
    if (do_deg && lane == 0) atomicAdd(deg + d, 1.0f);
}

// ---------------------------------------------------------------------------
// agg /= max(deg, 1)
// ---------------------------------------------------------------------------
__global__ void k_mean(float* __restrict__ agg, const float* __restrict__ deg,
                       long total) {
    long i = (long)blockIdx.x * blockDim.x + threadIdx.x;
    if (i >= total) return;
    agg[i] = agg[i] / fmaxf(deg[i >> 7], 1.0f);
}

// ---------------------------------------------------------------------------
// LayerNorm over D=128 per row, one wave/row (wave32, float4/lane),
// optional residual add then optional GELU:  out = act(LN(in) [+ res])
// ---------------------------------------------------------------------------
__global__ void __launch_bounds__(TPB)
k_ln(const float* __restrict__ in, const float* __restrict__ g,
     const float* __restrict__ b, const float* __restrict__ res,
     float* __restrict__ out, int rows, int do_gelu) {
    const int wave = threadIdx.x >> 5;
    const int lane = threadIdx.x & 31;
    const int row  = blockIdx.x * (blockDim.x >> 5) + wave;
    if (row >= rows) return;

    const float4 v = ((const float4*)(in + (long)row * D))[lane];
    float s  = v.x + v.y + v.z + v.w;
    float ss = v.x * v.x + v.y * v.y + v.z * v.z + v.w * v.w;
#pragma unroll
    for (int m = 16; m > 0; m >>= 1) {
        s  += __shfl_xor(s, m, 32);
        ss += __shfl_xor(ss, m, 32);
    }
    const float mu  = s * (1.0f / D);
    const float var = ss * (1.0f / D) - mu * mu;
    const float rs  = rsqrtf(var + 1e-5f);

    const float4 gg = ((const float4*)g)[lane];
    const float4 bb = ((const float4*)b)[lane];
    float4 o;
    o.x = (v.x - mu) * rs * gg.x + bb.x;
    o.y = (v.y - mu) * rs * gg.y + bb.y;
    o.z = (v.z - mu) * rs * gg.z + bb.z;
    o.w = (v.w - mu) * rs * gg.w + bb.w;
    if (res) {
        const float4 r = ((const float4*)(res + (long)row * D))[lane];
        o.x += r.x; o.y += r.y; o.z += r.z; o.w += r.w;
    }
    if (do_gelu) {
        o.x = gelu_exact(o.x); o.y = gelu_exact(o.y);
        o.z = gelu_exact(o.z); o.w = gelu_exact(o.w);
    }
    ((float4*)(out + (long)row * D))[lane] = o;
}

// ---------------------------------------------------------------------------
// Host-side orchestration
// ---------------------------------------------------------------------------
extern "C" void kernel_launch(void* const* d_in, const int* in_sizes, int n_in,
                              void* d_out, int out_size, void* d_ws,
                              size_t ws_size, hipStream_t stream) {
    const float* x     = (const float*)d_in[0];
    const int*   edges = (const int*)d_in[1];
    const float* dp_w  = (const float*)d_in[2];
    const float* dp_b  = (const float*)d_in[3];
    const float* sc_w  = (const float*)d_in[4];
    const float* sc_b  = (const float*)d_in[5];
    const float* g1_lw = (const float*)d_in[6];
    const float* g1_lb = (const float*)d_in[7];
    const float* g1_rw = (const float*)d_in[8];
    const float* n1_g  = (const float*)d_in[9];
    const float* n1_b  = (const float*)d_in[10];
    const float* g2_lw = (const float*)d_in[11];
    const float* g2_lb = (const float*)d_in[12];
    const float* g2_rw = (const float*)d_in[13];
    const float* n2_g  = (const float*)d_in[14];
    const float* n2_b  = (const float*)d_in[15];

    const int N = in_sizes[0] / D;
    const int E = in_sizes[1] / 2;
    const int* src = edges;
    const int* dst = edges + E;

    // workspace layout (floats): h | shortcut | agg | t | deg
    float* ws       = (float*)d_ws;
    const long ND   = (long)N * D;
    float* h        = ws;            // h0, later h1 (overwritten after use)
    float* shortcut = ws + ND;
    float* agg      = ws + 2 * ND;
    float* t        = ws + 3 * ND;
    float* deg      = ws + 4 * ND;

    float* outp = (float*)d_out;

    const int tiles      = N / 16;                       // 3125 (N % 16 == 0)
    const dim3 gGemm((tiles + WAVES - 1) / WAVES);       // 391 blocks
    const dim3 gEdge((E + (TPB / 32) - 1) / (TPB / 32)); // 1 wave/edge
    const dim3 gRow((N + (TPB / 32) - 1) / (TPB / 32));  // 1 wave/row
    const dim3 gElem((unsigned)((ND + TPB - 1) / TPB));
    const dim3 blk(TPB);

    // Stage 0: h = gelu(x @ dp_w^T + dp_b); shortcut = x @ sc_w^T + sc_b
    k_gemm<false><<<gGemm, blk, 0, stream>>>(x, dp_w, dp_b, nullptr, nullptr, h, N, 1);
    k_gemm<false><<<gGemm, blk, 0, stream>>>(x, sc_w, sc_b, nullptr, nullptr, shortcut, N, 0);

    // Stage 1: SAGE conv 1
    k_zero<<<1024, blk, 0, stream>>>(agg, ND);
    k_zero<<<64, blk, 0, stream>>>(deg, N);
    k_scatter<<<gEdge, blk, 0, stream>>>(h, src, dst, agg, deg, E, 1);
    k_mean<<<gElem, blk, 0, stream>>>(agg, deg, ND);
    k_gemm<true><<<gGemm, blk, 0, stream>>>(agg, g1_lw, g1_lb, h, g1_rw, t, N, 0);
    k_ln<<<gRow, blk, 0, stream>>>(t, n1_g, n1_b, nullptr, h, N, 1); // h := h1

    // Stage 2: SAGE conv 2 (degree reused)
    k_zero<<<1024, blk, 0, stream>>>(agg, ND);
    k_scatter<<<gEdge, blk, 0, stream>>>(h, src, dst, agg, deg, E, 0);
    k_mean<<<gElem, blk, 0, stream>>>(agg, deg, ND);
    k_gemm<true><<<gGemm, blk, 0, stream>>>(agg, g2_lw, g2_lb, h, g2_rw, t, N, 0);

    // Stage 3: out = gelu(LN(t) + shortcut)
    k_ln<<<gRow, blk, 0, stream>>>(t, n2_g, n2_b, shortcut, outp, N, 1);
}